// MultiCode_30185030156575
// MI455X (gfx1250) — compile-verified
//
#include <hip/hip_runtime.h>

#define TPB  256
#define ITER 32
#define LVLS 6
// total LDS floats: sum over lvl of (8<<lvl)*8 = 64*(2^6-1) = 4032  (16128 bytes)
#define LDS_FLOATS 4032

typedef float v4f __attribute__((ext_vector_type(4)));

__global__ __launch_bounds__(TPB) void interp_ngp1d_kernel(
    const float* __restrict__ t,
    const float* __restrict__ v0, const float* __restrict__ v1,
    const float* __restrict__ v2, const float* __restrict__ v3,
    const float* __restrict__ v4, const float* __restrict__ v5,
    float* __restrict__ out, int nChunks)
{
    __shared__ float smem[LDS_FLOATS];
    const float* vols[LVLS] = {v0, v1, v2, v3, v4, v5};
    const int tid = threadIdx.x;

    // ---- Stage all 6 tables into LDS via async global->LDS B128 DMA ----
    // level lvl: (16<<lvl) float4 chunks, LDS float4 base = 16*(2^lvl - 1)
#pragma unroll
    for (int lvl = 0; lvl < LVLS; ++lvl) {
        const int n4    = 16 << lvl;
        const int base4 = 16 * ((1 << lvl) - 1);
        const v4f* src  = (const v4f*)vols[lvl];
        for (int i = tid; i < n4; i += TPB) {
            // low 32 bits of a generic LDS pointer == LDS byte offset (aperture form)
            unsigned lds_off = (unsigned)(unsigned long long)(uintptr_t)&smem[(base4 + i) * 4];
            unsigned long long ga = (unsigned long long)(uintptr_t)(src + i);
            asm volatile("global_load_async_to_lds_b128 %0, %1, off"
                         :: "v"(lds_off), "v"(ga) : "memory");
        }
    }
    asm volatile("s_wait_asynccnt 0" ::: "memory");
    __syncthreads();

    // ---- Each thread produces ITER float4 output chunks ----
    // chunk c -> point n = c/12, k = c%12, level = k>>1, channel-half = k&1
    int chunk = blockIdx.x * (TPB * ITER) + tid;
    int n = chunk / 12;              // one division; thereafter incremental
    int k = chunk - n * 12;          // 256 = 21*12 + 4  per-iteration stride

#pragma unroll 4
    for (int it = 0; it < ITER; ++it) {
        if (chunk < nChunks) {
            const int lvl  = k >> 1;
            const int half = k & 1;

            float tc = fminf(fmaxf(t[n], 0.0f), 1.0f);
            const int size = 8 << lvl;
            const float pos = tc * (float)(size - 1);
            int i0 = (int)pos;                 // pos >= 0, trunc == floor
            i0 = (i0 > size - 2) ? (size - 2) : i0;
            const float w  = pos - (float)i0;
            const float iw = 1.0f - w;

            const int base = 64 * ((1 << lvl) - 1) + i0 * 8 + half * 4;
            const v4f a = *(const v4f*)&smem[base];      // row i0
            const v4f b = *(const v4f*)&smem[base + 8];  // row i0+1

            v4f r;
            r.x = a.x * iw + b.x * w;
            r.y = a.y * iw + b.y * w;
            r.z = a.z * iw + b.z * w;
            r.w = a.w * iw + b.w * w;

            // streaming write-once output: non-temporal, contiguous 512B/wave
            __builtin_nontemporal_store(r, (v4f*)out + chunk);
        }
        chunk += TPB;
        n += 21;
        k += 4;
        if (k >= 12) { k -= 12; ++n; }
    }
}

extern "C" void kernel_launch(void* const* d_in, const int* in_sizes, int n_in,
                              void* d_out, int out_size, void* d_ws, size_t ws_size,
                              hipStream_t stream) {
    (void)n_in; (void)out_size; (void)d_ws; (void)ws_size;
    const int npoints = in_sizes[0];          // N = 2,000,000
    const int nChunks = npoints * 12;         // 12 float4 chunks per point (48 floats)
    const int perBlock = TPB * ITER;
    const int blocks = (nChunks + perBlock - 1) / perBlock;

    interp_ngp1d_kernel<<<blocks, TPB, 0, stream>>>(
        (const float*)d_in[0],
        (const float*)d_in[1], (const float*)d_in[2], (const float*)d_in[3],
        (const float*)d_in[4], (const float*)d_in[5], (const float*)d_in[6],
        (float*)d_out, nChunks);
}